// GMaxP_41592463294652
// MI455X (gfx1250) — compile-verified
//
#include <hip/hip_runtime.h>
#include <math.h>

// Problem constants (from reference): b=32, n=w*h=1024, d=512, emb=512, lambda=1
#define B_    32
#define N_    1024
#define D_    512
#define EMB_  512
#define LAM_  1.0f

typedef __attribute__((ext_vector_type(2))) float v2f;
typedef __attribute__((ext_vector_type(8))) float v8f;

// ---------------------------------------------------------------------------
// Kernel 1: M[b] = P[b]^T P[b]   (P = x reshaped to N_ x D_); +lambda*I is
// applied later in chol_solve (which rewrites the diagonal anyway).
//
// Register-tiled fp32 WMMA 16x16x4:
//   - block = 128 threads = 4 waves; block tile = 64x64 of G
//   - each wave computes a 32x32 quadrant = 2x2 grid of 16x16 WMMA tiles
//   - per K-step of 4: 8 loads -> 4 WMMAs (2x the FLOP/byte of 1x1 tiling)
//
// Fragment layouts (ISA 7.12.2, 32-bit data, wave32):
//   A 16x4 (MxK): lane L holds col i0+(L&15), rows k+2*(L>=16)+{0,1}
//   B 4x16 (KxN): same pattern with col j0+(L&15)
//   C/D 16x16:    VGPR v -> row v + 8*(L>=16), col L&15
// ---------------------------------------------------------------------------
__global__ __launch_bounds__(128) void gram_wmma(const float* __restrict__ x,
                                                 float* __restrict__ M) {
  const int wave = threadIdx.x >> 5;     // 0..3
  const int lane = threadIdx.x & 31;
  const int wi   = wave >> 1;            // quadrant row
  const int wj   = wave & 1;             // quadrant col
  const int b    = blockIdx.z;
  const int i0   = blockIdx.y * 64 + wi * 32;   // G row base (32-wide)
  const int j0   = blockIdx.x * 64 + wj * 32;   // G col base (32-wide)
  const int m    = lane & 15;
  const int kk   = (lane >> 4) << 1;     // 0 or 2

  const float* __restrict__ P = x + (size_t)b * N_ * D_;
  const float* pa0 = P + i0 + m;         // A frag, cols i0   .. i0+15
  const float* pa1 = pa0 + 16;           // A frag, cols i0+16.. i0+31
  const float* pb0 = P + j0 + m;         // B frag, cols j0   .. j0+15
  const float* pb1 = pb0 + 16;           // B frag, cols j0+16.. j0+31

  v8f c00 = {}, c01 = {}, c10 = {}, c11 = {};
#pragma unroll 2
  for (int k = 0; k < N_; k += 4) {
    const size_t r0 = (size_t)(k + kk) * D_;
    const size_t r1 = r0 + D_;
    v2f a0, a1, b0, b1;
    a0.x = pa0[r0]; a0.y = pa0[r1];
    a1.x = pa1[r0]; a1.y = pa1[r1];
    b0.x = pb0[r0]; b0.y = pb0[r1];
    b1.x = pb1[r0]; b1.y = pb1[r1];
    // D = A x B + C : (neg_a, A, neg_b, B, c_mod, C, reuse_a, reuse_b)
    c00 = __builtin_amdgcn_wmma_f32_16x16x4_f32(false, a0, false, b0, (short)0, c00, false, false);
    c01 = __builtin_amdgcn_wmma_f32_16x16x4_f32(false, a0, false, b1, (short)0, c01, false, false);
    c10 = __builtin_amdgcn_wmma_f32_16x16x4_f32(false, a1, false, b0, (short)0, c10, false, false);
    c11 = __builtin_amdgcn_wmma_f32_16x16x4_f32(false, a1, false, b1, (short)0, c11, false, false);
  }

  float* __restrict__ Mb = M + (size_t)b * D_ * D_;
  const int rb = (lane >> 4) << 3;       // 0 or 8
  const int cA = j0 + m, cB = cA + 16;
#pragma unroll
  for (int v = 0; v < 8; ++v) {
    const int rA = i0 + rb + v;          // rows of c00/c01
    const int rB = rA + 16;              // rows of c10/c11
    Mb[(size_t)rA * D_ + cA] = c00[v];
    Mb[(size_t)rA * D_ + cB] = c01[v];
    Mb[(size_t)rB * D_ + cA] = c10[v];
    Mb[(size_t)rB * D_ + cB] = c11[v];
  }
}

// ---------------------------------------------------------------------------
// Kernel 2: q[b][j] = sum_k P[b][k][j]   (q = P^T * ones)
// ---------------------------------------------------------------------------
__global__ __launch_bounds__(256) void colsum(const float* __restrict__ x,
                                              float* __restrict__ q) {
  const int idx = blockIdx.x * 256 + threadIdx.x;   // b*D_ + j
  const int b = idx / D_, j = idx - b * D_;
  const float* __restrict__ P = x + (size_t)b * N_ * D_;
  float s = 0.f;
  for (int k = 0; k < N_; ++k) s += P[(size_t)k * D_ + j];
  q[idx] = s;
}

// ---------------------------------------------------------------------------
// Kernel 3: per batch: save G diag, form M = G + lambda*I on the diagonal,
// Cholesky (lower triangle in place, strict upper left as original G), then
// solve (I+G) y = q via L L^T. One workgroup per batch; pivot column in LDS.
// ---------------------------------------------------------------------------
__global__ __launch_bounds__(256) void chol_solve(float* __restrict__ M,
                                                  const float* __restrict__ q,
                                                  float* __restrict__ y,
                                                  float* __restrict__ dsave) {
  const int b = blockIdx.x, tid = threadIdx.x, nt = blockDim.x;
  const int n = D_;
  float* __restrict__ A = M + (size_t)b * n * n;

  __shared__ float z[D_];
  __shared__ float colk[D_];
  __shared__ float s_piv;

  for (int i = tid; i < n; i += nt) {
    const float g = A[(size_t)i * n + i];
    dsave[b * n + i] = g;                  // pure G diagonal (for finalize)
    A[(size_t)i * n + i] = g + LAM_;       // M = G + lambda*I
    z[i] = q[b * n + i];
  }
  __syncthreads();

  // Right-looking Cholesky, lower triangle only.
  for (int k = 0; k < n; ++k) {
    if (tid == 0) {
      const float p_ = sqrtf(A[(size_t)k * n + k]);
      A[(size_t)k * n + k] = p_;
      s_piv = p_;
    }
    __syncthreads();
    const float inv = 1.0f / s_piv;
    for (int i = k + 1 + tid; i < n; i += nt) {
      const float v_ = A[(size_t)i * n + k] * inv;
      A[(size_t)i * n + k] = v_;
      colk[i] = v_;
    }
    __syncthreads();
    for (int i = k + 1 + tid; i < n; i += nt) {
      const float lik = colk[i];
      float* Ai = A + (size_t)i * n;
      for (int j = k + 1; j <= i; ++j) Ai[j] -= lik * colk[j];  // contiguous row
    }
    __syncthreads();
  }

  // Forward solve L z = q
  for (int k = 0; k < n; ++k) {
    if (tid == 0) z[k] /= A[(size_t)k * n + k];
    __syncthreads();
    const float zk = z[k];
    for (int i = k + 1 + tid; i < n; i += nt) z[i] -= A[(size_t)i * n + k] * zk;
    __syncthreads();
  }
  // Backward solve L^T y = z (in place)
  for (int k = n - 1; k >= 0; --k) {
    if (tid == 0) z[k] /= A[(size_t)k * n + k];
    __syncthreads();
    const float yk = z[k];
    const float* Ak = A + (size_t)k * n;
    for (int i = tid; i < k; i += nt) z[i] -= Ak[i] * yk;       // contiguous row
    __syncthreads();
  }
  for (int i = tid; i < n; i += nt) y[b * n + i] = z[i];
}

// ---------------------------------------------------------------------------
// Kernel 4: pooled = q - G y  (G rebuilt from intact strict-upper triangle +
// saved diag), out = pooled @ W, L2-normalize (eps=1e-12, rsqrt).
// One workgroup (512 threads) per batch; thread j owns output element j.
// ---------------------------------------------------------------------------
__global__ __launch_bounds__(512) void finalize(const float* __restrict__ M,
                                                const float* __restrict__ dsave,
                                                const float* __restrict__ q,
                                                const float* __restrict__ y,
                                                const float* __restrict__ W,
                                                float* __restrict__ out) {
  const int b = blockIdx.x, j = threadIdx.x;
  const int n = D_;
  const float* __restrict__ A = M + (size_t)b * n * n;

  __shared__ float ys[D_];
  __shared__ float pooled[D_];
  __shared__ float red[D_];

  ys[j] = y[b * n + j];
  __syncthreads();

  float acc = 0.f;
  const float* Aj = A + (size_t)j * n;
  for (int k = 0; k < n; ++k) {
    float g;
    if (k > j)      g = Aj[k];                        // strict upper: original G
    else if (k < j) g = A[(size_t)k * n + j];         // symmetry via upper
    else            g = dsave[b * n + j];             // saved pure-G diagonal
    acc += g * ys[k];
  }
  pooled[j] = q[b * n + j] - acc;
  __syncthreads();

  float o = 0.f;
  for (int d = 0; d < D_; ++d) o += pooled[d] * W[(size_t)d * EMB_ + j];

  red[j] = o * o;
  __syncthreads();
  for (int s = 256; s > 0; s >>= 1) {
    if (j < s) red[j] += red[j + s];
    __syncthreads();
  }
  const float scale = rsqrtf(fmaxf(red[0], 1e-12f));
  out[b * EMB_ + j] = o * scale;
}

// ---------------------------------------------------------------------------
// Workspace layout (needs ~33.7 MB; L2 is 192 MB so M stays cache-resident):
//   M     : B_*D_*D_ floats (32 MB)  G, then Cholesky L in lower triangle
//   q     : B_*D_ floats
//   y     : B_*D_ floats
//   dsave : B_*D_ floats
// ---------------------------------------------------------------------------
extern "C" void kernel_launch(void* const* d_in, const int* in_sizes, int n_in,
                              void* d_out, int out_size, void* d_ws, size_t ws_size,
                              hipStream_t stream) {
  const float* x = (const float*)d_in[0];   // (32,32,32,512) fp32
  const float* W = (const float*)d_in[1];   // (512,512) fp32
  float* out = (float*)d_out;               // (32,512) fp32

  char* ws = (char*)d_ws;
  float* M  = (float*)ws;
  float* q  = (float*)(ws + (size_t)B_ * D_ * D_ * sizeof(float));
  float* y  = q + B_ * D_;
  float* ds = y + B_ * D_;

  gram_wmma<<<dim3(D_ / 64, D_ / 64, B_), 128, 0, stream>>>(x, M);
  colsum<<<(B_ * D_) / 256, 256, 0, stream>>>(x, q);
  chol_solve<<<B_, 256, 0, stream>>>(M, q, y, ds);
  finalize<<<B_, 512, 0, stream>>>(M, ds, q, y, W, out);
}